// ExponentialMovingAverage_28183575396436
// MI455X (gfx1250) — compile-verified
//
#include <hip/hip_runtime.h>
#include <stdint.h>

// Problem constants (fixed by the reference).
#define B_SZ   16
#define T_SZ   8192
#define C_SZ   256
#define L_CH   128                 // timesteps per chunk (per block)
#define K_CH   (T_SZ / L_CH)       // 64 chunks
#define R_ST   8                   // rows (timesteps) staged per async step
#define NSTAGE (L_CH / R_ST)       // 16 double-buffered stages per chunk

// ---------------------------------------------------------------------------
// CDNA5 async global->LDS helpers (gfx1250). The async engine writes LDS and
// completion is tracked per-wave in ASYNCcnt; loads complete IN ORDER, so
// "s_wait_asynccnt N" means "all but the newest N issues have landed".
// ---------------------------------------------------------------------------
__device__ __forceinline__ uint32_t lds_addr_of(const void* p) {
    // Generic pointers to __shared__ carry the LDS byte offset in the low
    // 32 bits (LDS aperture); that is exactly what the async VDST VGPR wants.
    return (uint32_t)(uintptr_t)p;
}

__device__ __forceinline__ void async_load_b128(uint32_t lds_off, const float* g) {
    asm volatile("global_load_async_to_lds_b128 %0, %1, off"
                 :: "v"(lds_off), "v"(g)
                 : "memory");
}

template <int N>
__device__ __forceinline__ void wait_asynccnt() {
    asm volatile("s_wait_asynccnt %0" :: "i"(N) : "memory");
}

// ---------------------------------------------------------------------------
// Pass 1 / Pass 3 (shared body). Each block owns chunk (b = blockIdx.y,
// k = blockIdx.x): 128 timesteps x 256 channels. Thread tid = channel.
// Input is staged through LDS with double-buffered async b128 copies:
// each stage moves 8 rows = 8KB; each of the 8 waves issues exactly 2
// global_load_async_to_lds_b128 (32 lanes x 16B x 2 = 1KB ... 8 waves = 8KB).
// WRITE_OUT=false : start from 0, emit only the chunk carry-sum B_k.
// WRITE_OUT=true  : start from carry h_k, emit all 128 outputs (NT stores so
//                   the 134MB output stream does not evict x from the 192MB L2).
// ---------------------------------------------------------------------------
template <bool WRITE_OUT>
__global__ __launch_bounds__(C_SZ)
void ema_chunk_kernel(const float* __restrict__ x,       // [B,T,C]
                      const float* __restrict__ smooth,  // [C]
                      const float* __restrict__ carry,   // [B,K,C] (pass 3 only)
                      float* __restrict__ dst)           // pass1: Bk [B,K,C]; pass3: y [B,T,C]
{
    __shared__ float lds[2][R_ST * C_SZ];   // 2 x 8KB double buffer

    const int k   = blockIdx.x;
    const int b   = blockIdx.y;
    const int tid = threadIdx.x;            // channel index

    const float w = fminf(fmaxf(smooth[tid], 0.0f), 1.0f);
    const float a = 1.0f - w;

    const float* gx = x + ((size_t)b * T_SZ + (size_t)k * L_CH) * C_SZ;

    // Prologue: stage 0 -> buffer 0.
    {
        const uint32_t l0 = lds_addr_of(&lds[0][0]) + (uint32_t)tid * 16u;
        async_load_b128(l0,          gx + (size_t)tid * 4);
        async_load_b128(l0 + 4096u,  gx + 1024 + (size_t)tid * 4);
    }

    float y = WRITE_OUT ? carry[((size_t)b * K_CH + k) * C_SZ + tid] : 0.0f;

    for (int s = 0; s < NSTAGE; ++s) {
        const int p = s & 1;

        if (s + 1 < NSTAGE) {
            // Prefetch next stage into the other buffer, then wait for the
            // 2 older issues (current buffer) to land: asynccnt <= 2.
            const float* src = gx + (size_t)(s + 1) * R_ST * C_SZ;
            const uint32_t l1 = lds_addr_of(&lds[p ^ 1][0]) + (uint32_t)tid * 16u;
            async_load_b128(l1,         src + (size_t)tid * 4);
            async_load_b128(l1 + 4096u, src + 1024 + (size_t)tid * 4);
            wait_asynccnt<2>();
        } else {
            wait_asynccnt<0>();
        }
        __syncthreads();   // all waves' slices of buffer p are now in LDS

        #pragma unroll
        for (int j = 0; j < R_ST; ++j) {
            const float xv = lds[p][j * C_SZ + tid];
            y = fmaf(a, y, w * xv);
            if (WRITE_OUT) {
                const size_t t = (size_t)k * L_CH + (size_t)s * R_ST + j;
                __builtin_nontemporal_store(y, dst + ((size_t)b * T_SZ + t) * C_SZ + tid);
            }
        }
        __syncthreads();   // buffer p free for reuse next iteration
    }

    if (!WRITE_OUT) {
        // Final local value == B_k = sum_i a^(L-1-i) * w * x_i
        dst[((size_t)b * K_CH + k) * C_SZ + tid] = y;
    }
}

// ---------------------------------------------------------------------------
// Pass 2: in-place carry scan over the K=64 chunk summaries.
// BkH[b,k,c] holds B_k on entry; on exit it holds h_k (carry-in state for
// chunk k): h_0 = initial_state, h_{k+1} = a^128 * h_k + B_k.
// Loads are h-independent so the unrolled loop pipelines them deeply.
// ---------------------------------------------------------------------------
__global__ __launch_bounds__(C_SZ)
void ema_carry_scan(float* __restrict__ BkH,            // [B,K,C] in/out
                    const float* __restrict__ init,     // [B,C]
                    const float* __restrict__ smooth)   // [C]
{
    const int b = blockIdx.x;
    const int c = threadIdx.x;

    const float w = fminf(fmaxf(smooth[c], 0.0f), 1.0f);
    float aL = 1.0f - w;
    #pragma unroll
    for (int i = 0; i < 7; ++i) aL *= aL;   // a^128

    float h = init[(size_t)b * C_SZ + c];

    #pragma unroll 8
    for (int kk = 0; kk < K_CH; ++kk) {
        const size_t idx = ((size_t)b * K_CH + kk) * C_SZ + c;
        const float bk = BkH[idx];   // read B_k
        BkH[idx] = h;                // overwrite with carry-in h_k
        h = fmaf(aL, h, bk);
    }
}

// ---------------------------------------------------------------------------
// Host launcher. Inputs per reference order: inputs [B,T,C] f32,
// initial_state [B,C] f32, smooth [C] f32. Output y [B,T,C] f32.
// Workspace: 1 MB (B*K*C floats), fully rewritten each call (deterministic).
// ---------------------------------------------------------------------------
extern "C" void kernel_launch(void* const* d_in, const int* in_sizes, int n_in,
                              void* d_out, int out_size, void* d_ws, size_t ws_size,
                              hipStream_t stream) {
    (void)in_sizes; (void)n_in; (void)out_size; (void)ws_size;

    const float* x      = (const float*)d_in[0];
    const float* init   = (const float*)d_in[1];
    const float* smooth = (const float*)d_in[2];
    float*       out    = (float*)d_out;
    float*       BkH    = (float*)d_ws;     // B*K*C floats = 1 MB

    dim3 grid(K_CH, B_SZ);   // 64 x 16 = 1024 blocks, 8 waves each

    // Pass 1: chunk summaries (also warms x into the 192MB L2).
    ema_chunk_kernel<false><<<grid, C_SZ, 0, stream>>>(x, smooth, nullptr, BkH);
    // Pass 2: tiny in-place carry scan.
    ema_carry_scan<<<B_SZ, C_SZ, 0, stream>>>(BkH, init, smooth);
    // Pass 3: final outputs; x re-read (L2-resident), y written non-temporal.
    ema_chunk_kernel<true><<<grid, C_SZ, 0, stream>>>(x, smooth, BkH, out);
}